// ESpaceLoss_59734405153030
// MI455X (gfx1250) — compile-verified
//
#include <hip/hip_runtime.h>
#include <math.h>

// ---------------------------------------------------------------------------
// ESpaceLoss for MI455X (gfx1250, wave32)
//
// Phase 1 (blocked-scan IIR, fp64):
//   A) per-(recurrence,chunk) zero-state biquad over L=1000 samples -> final
//      2-state;  B) per-recurrence serial fix-up with M = companion^L
//      (binary exponentiation);  C) re-run only the valid chunks (40..79)
//      with correct initial states.  Critical path: ~2L+80 steps vs 120000.
// Phase 2: tanh(W @ X) via V_WMMA_F32_16X16X4_F32 (exact fp32 matrix path),
//          branchless fast tanh (v_exp_f32 + v_rcp_f32), fused |pred-target|
//          partial reduction.
// Phase 3: deterministic final reduction -> scalar mean.
// ---------------------------------------------------------------------------

typedef float v2f __attribute__((ext_vector_type(2)));
typedef float v8f __attribute__((ext_vector_type(8)));

#define T_LEN    120000
#define MARGIN   40000
#define TV       40000          // valid samples per batch
#define NCH      128            // 2*RES_SIZE probe channels
#define NBATCH   4
#define NREC     1024           // 2 signals * 4 batch * 128 channels

#define L_CHUNK  1000           // chunk length (MARGIN % L_CHUNK == 0)
#define N_CHUNK_A 80            // chunks 0..79 (states feeding valid region)
#define C_FIRST   40            // first valid chunk (t = 40000)
#define N_CHUNK_C 40            // chunks 40..79 cover [40000, 80000)

#define TTILE    32             // time columns per block in phase 2
#define NTBLK    (TV / TTILE)   // 1250

// ---------------------------------------------------------------------------
// Phase 1 A/C: one block = (signal*batch, chunk); 128 threads = 128
// recurrences (dir = ch>>6, resonator = ch&63).  MODE 0: zero-state, emit
// final states.  MODE 1: init from fixed-up states, emit probe outputs.
// ---------------------------------------------------------------------------
template <int MODE>
__global__ __launch_bounds__(128) void espace_biquad_chunk_kernel(
    const float* __restrict__ pred, const float* __restrict__ tgt,
    const double* __restrict__ a1, const double* __restrict__ a2,
    const double* __restrict__ b0,
    double* __restrict__ S0,          // [NREC][N_CHUNK_A][2] zero-state finals
    const double* __restrict__ V,     // [NREC][N_CHUNK_A][2] fixed-up inits
    float* __restrict__ X,            // [2][4][TV][NCH]
    int chunk_base)
{
    __shared__ float sF[L_CHUNK];     // forward window
    __shared__ float sB[L_CHUNK];     // time-reversed window
    const int sb    = blockIdx.y;             // s*4 + b, 0..7
    const int s     = sb >> 2;
    const int b     = sb & 3;
    const int chunk = chunk_base + blockIdx.x;
    const int t0    = chunk * L_CHUNK;
    const int ch    = threadIdx.x;            // 0..127
    const int dir   = ch >> 6;                // 0 = fwd, 1 = bwd
    const int c     = ch & 63;

    const float* x = (s ? tgt : pred) + (size_t)b * T_LEN;
    for (int j = ch; j < L_CHUNK; j += 128) {
        sF[j] = x[t0 + j];
        sB[j] = x[T_LEN - 1 - (t0 + j)];
    }
    __syncthreads();

    const double A1 = a1[c], A2 = a2[c], B0 = b0[c];
    const size_t rec = (size_t)sb * 128 + ch;

    double y1, y2;
    if (MODE == 0) {
        y1 = 0.0; y2 = 0.0;
    } else {
        const double* v = V + (rec * N_CHUNK_A + chunk) * 2;
        y1 = v[0]; y2 = v[1];
    }

    float* Xo = X + (size_t)sb * TV * NCH + ch;

    for (int i = 0; i < L_CHUNK; ++i) {
        const double xt = (double)(dir ? sB[i] : sF[i]);
        // y[t] = b0*x - a1*y[t-1] - a2*y[t-2]; inner fma depends on y[t-2]
        // (ready one step early) -> critical path ~1 fp64 FMA per step.
        const double p = fma(-A2, y2, B0 * xt);
        const double y = fma(-A1, y1, p);
        y2 = y1; y1 = y;
        if (MODE == 1) {
            const int t   = t0 + i;
            const int idx = dir ? (T_LEN - 1 - t) : t;   // in [MARGIN, T-MARGIN)
            Xo[(size_t)(idx - MARGIN) * NCH] = (float)y;
        }
    }

    if (MODE == 0) {
        double* sp = S0 + (rec * N_CHUNK_A + chunk) * 2;
        sp[0] = y1; sp[1] = y2;   // state entering next chunk
    }
}

// ---------------------------------------------------------------------------
// Phase 1 B: per-recurrence fix-up.  M = A^L with A = [[-a1,-a2],[1,0]]
// (|eig| = r < 1 -> numerically benign), then v_in[i+1] = M v_in[i] + S0[i].
// ---------------------------------------------------------------------------
__global__ __launch_bounds__(256) void espace_fixup_kernel(
    const double* __restrict__ a1, const double* __restrict__ a2,
    const double* __restrict__ S0, double* __restrict__ V)
{
    const int rec = blockIdx.x * 256 + threadIdx.x;   // 0..1023
    const int c   = rec & 63;
    const double A1 = a1[c], A2 = a2[c];

    // M = A^L_CHUNK by binary exponentiation (2x2, fp64)
    double m00 = 1.0, m01 = 0.0, m10 = 0.0, m11 = 1.0;
    double b00 = -A1, b01 = -A2, b10 = 1.0, b11 = 0.0;
    int e = L_CHUNK;
    while (e) {
        if (e & 1) {
            const double n00 = m00*b00 + m01*b10;
            const double n01 = m00*b01 + m01*b11;
            const double n10 = m10*b00 + m11*b10;
            const double n11 = m10*b01 + m11*b11;
            m00 = n00; m01 = n01; m10 = n10; m11 = n11;
        }
        e >>= 1;
        if (e) {
            const double c00 = b00*b00 + b01*b10;
            const double c01 = b00*b01 + b01*b11;
            const double c10 = b10*b00 + b11*b10;
            const double c11 = b10*b01 + b11*b11;
            b00 = c00; b01 = c01; b10 = c10; b11 = c11;
        }
    }

    const double* s0 = S0 + (size_t)rec * N_CHUNK_A * 2;
    double*       vv = V  + (size_t)rec * N_CHUNK_A * 2;
    double v0 = 0.0, v1 = 0.0;                 // state entering chunk 0
    for (int i = 0; i < N_CHUNK_A; ++i) {
        vv[2*i]   = v0;
        vv[2*i+1] = v1;
        const double n0 = m00*v0 + m01*v1 + s0[2*i];
        const double n1 = m10*v0 + m11*v1 + s0[2*i+1];
        v0 = n0; v1 = n1;
    }
}

// ---------------------------------------------------------------------------
// branchless fast tanh: v_exp_f32 + v_rcp_f32, no EXEC divergence, no
// IEEE division ladder.  tanh(x) = sign(x) * (1-e)/(1+e), e = exp(-2|x|);
// denominator in [1,2] so v_rcp_f32's ~1ulp error is harmless.
// ---------------------------------------------------------------------------
__device__ __forceinline__ float fast_tanhf(float x) {
    const float e = __expf(-2.0f * fabsf(x));
    const float t = (1.0f - e) * __builtin_amdgcn_rcpf(1.0f + e);
    return copysignf(t, x);
}

// ---------------------------------------------------------------------------
// Phase 2: one block = (batch, 32 time cols); 8 waves, each wave owns one
// 16(M) x 16(N) output tile and K-steps 32 x V_WMMA_F32_16X16X4_F32 for BOTH
// signals (A operand reused).  Fused tanh + |diff| + deterministic reduce.
// ---------------------------------------------------------------------------
__global__ __launch_bounds__(256) void espace_tank_loss_kernel(
    const float* __restrict__ W, const float* __restrict__ X,
    float* __restrict__ partial)
{
    __shared__ float wsum[8];
    const int bb    = blockIdx.x / NTBLK;     // batch 0..3
    const int tblk  = blockIdx.x % NTBLK;     // time block 0..1249
    const int w     = threadIdx.x >> 5;       // wave 0..7
    const int l     = threadIdx.x & 31;       // lane
    const int otile = w >> 1;                 // output-row tile 0..3
    const int ttile = w & 1;                  // time tile 0..1
    const int lo    = l & 15;
    const int koff  = (l >> 4) * 2;           // lane half selects K {0,1}/{2,3}

    const int m    = otile * 16 + lo;                    // A row (out chan)
    const int tcol = tblk * TTILE + ttile * 16 + lo;     // B col (time)

    const float* Arow = W + (size_t)m * NCH;
    const float* Bp   = X + ((size_t)(0 * NBATCH + bb) * TV + tcol) * NCH;
    const float* Bt   = X + ((size_t)(1 * NBATCH + bb) * TV + tcol) * NCH;

    v8f accP = {0.f,0.f,0.f,0.f,0.f,0.f,0.f,0.f};
    v8f accT = {0.f,0.f,0.f,0.f,0.f,0.f,0.f,0.f};

    #pragma unroll 4
    for (int k = 0; k < NCH; k += 4) {
        const v2f a  = *(const v2f*)(Arow + k + koff);
        const v2f bp = *(const v2f*)(Bp   + k + koff);
        const v2f bt = *(const v2f*)(Bt   + k + koff);
        accP = __builtin_amdgcn_wmma_f32_16x16x4_f32(
                   false, a, false, bp, (short)0, accP, false, false);
        accT = __builtin_amdgcn_wmma_f32_16x16x4_f32(
                   false, a, false, bt, (short)0, accT, false, false);
    }

    float sum = 0.f;
    #pragma unroll
    for (int r = 0; r < 8; ++r)
        sum += fabsf(fast_tanhf(accP[r]) - fast_tanhf(accT[r]));

    // wave32 butterfly reduce
    #pragma unroll
    for (int off = 16; off > 0; off >>= 1)
        sum += __shfl_xor(sum, off, 32);

    if (l == 0) wsum[w] = sum;
    __syncthreads();
    if (threadIdx.x == 0) {
        float tot = 0.f;
        #pragma unroll
        for (int i = 0; i < 8; ++i) tot += wsum[i];
        partial[blockIdx.x] = tot;
    }
}

// ---------------------------------------------------------------------------
// Phase 3: deterministic final reduction -> mean
// ---------------------------------------------------------------------------
__global__ __launch_bounds__(256) void espace_reduce_kernel(
    const float* __restrict__ partial, int n, float* __restrict__ out,
    float inv_n)
{
    __shared__ float sm[256];
    float s = 0.f;
    for (int i = threadIdx.x; i < n; i += 256) s += partial[i];
    sm[threadIdx.x] = s;
    __syncthreads();
    for (int st = 128; st > 0; st >>= 1) {
        if (threadIdx.x < st) sm[threadIdx.x] += sm[threadIdx.x + st];
        __syncthreads();
    }
    if (threadIdx.x == 0) out[0] = sm[0] * inv_n;
}

// ---------------------------------------------------------------------------
extern "C" void kernel_launch(void* const* d_in, const int* in_sizes, int n_in,
                              void* d_out, int out_size, void* d_ws, size_t ws_size,
                              hipStream_t stream)
{
    (void)in_sizes; (void)n_in; (void)out_size; (void)ws_size;
    const float*  pred = (const float*) d_in[0];   // [4,1,120000] f32
    const float*  tgt  = (const float*) d_in[1];   // [4,1,120000] f32
    const double* a1   = (const double*)d_in[2];   // [64] f64
    const double* a2   = (const double*)d_in[3];   // [64] f64
    const double* b0   = (const double*)d_in[4];   // [64] f64
    const float*  W    = (const float*) d_in[5];   // [64,128] f32

    // workspace layout
    float* X       = (float*)d_ws;                         // ~164 MB (L2-resident)
    float* partial = X + (size_t)2 * NBATCH * TV * NCH;    // 5000 floats
    size_t off = ((size_t)2 * NBATCH * TV * NCH + NBATCH * NTBLK) * sizeof(float);
    off = (off + 15) & ~(size_t)15;
    double* S0 = (double*)((char*)d_ws + off);             // [NREC][80][2]
    double* V  = S0 + (size_t)NREC * N_CHUNK_A * 2;        // [NREC][80][2]

    // Phase 1A: zero-state chunk scan (chunks 0..79)
    espace_biquad_chunk_kernel<0><<<dim3(N_CHUNK_A, 8), 128, 0, stream>>>(
        pred, tgt, a1, a2, b0, S0, V, X, 0);
    // Phase 1B: serial inter-chunk state fix-up (80 tiny steps)
    espace_fixup_kernel<<<NREC / 256, 256, 0, stream>>>(a1, a2, S0, V);
    // Phase 1C: re-run valid chunks 40..79 with correct initial states
    espace_biquad_chunk_kernel<1><<<dim3(N_CHUNK_C, 8), 128, 0, stream>>>(
        pred, tgt, a1, a2, b0, S0, V, X, C_FIRST);

    // Phase 2: WMMA tank + fused loss partials
    const int nblk = NBATCH * NTBLK;   // 5000
    espace_tank_loss_kernel<<<nblk, 256, 0, stream>>>(W, X, partial);

    // Phase 3: deterministic mean
    const float inv_n = 1.0f / (4.0f * 64.0f * 40000.0f);
    espace_reduce_kernel<<<1, 256, 0, stream>>>(partial, nblk,
                                                (float*)d_out, inv_n);
}